// EdgeDecoder_11802570129872
// MI455X (gfx1250) — compile-verified
//
#include <hip/hip_runtime.h>

// EdgeDecoder on MI455X (gfx1250):
//   out[e] = dot(x_src[i0[e]], w[:128]) + dot(x_dst[i1[e]], w[128:]) + bias
//
// Strategy: factor the shared weight out of the 2M-edge gather.
//   Pass 1 (WMMA GEMV): s_src[n] = x_src[n].w_src, s_dst[n] = x_dst[n].w_dst
//     -> streams 102 MB once, emits v_wmma_f32_16x16x4_f32 (16-row tiles / wave)
//   Pass 2: out[e] = s_src[i0] + s_dst[i1] + bias  (400KB tables, L2-resident)
// Total HBM traffic ~130 MB  =>  ~6 us at 23.3 TB/s, vs ~2 GB naive.

typedef __attribute__((ext_vector_type(2))) float v2f;
typedef __attribute__((ext_vector_type(8))) float v8f;

#define HIDDEN 128

// One wave32 per 16-node tile. A-matrix (16x4 f32): lane L holds row M=L&15,
// K = {kp, kp+1} with kp = (L>=16)?2:0  -> contiguous float2 load per lane.
// B-matrix (4x16 f32): N = L&15 striped over lanes, K split the same way;
// only column N==0 carries the weight vector. After 32 K-chunks, column 0 of
// the 16x16 f32 accumulator (lanes 0 and 16, acc VGPRs 0..7) holds the dots.
__global__ __launch_bounds__(256) void gemv_precompute(
    const float* __restrict__ x_src,
    const float* __restrict__ x_dst,
    const float* __restrict__ weight,   // [2*HIDDEN]
    float* __restrict__ s_out,          // [2*n_nodes] workspace
    int n_nodes, int tiles_per_table)
{
  const int lane = threadIdx.x & 31;
  const int tile = blockIdx.x * (blockDim.x >> 5) + (threadIdx.x >> 5);
  if (tile >= 2 * tiles_per_table) return;   // whole wave exits together

  const int table    = (tile >= tiles_per_table) ? 1 : 0;
  const int t        = tile - table * tiles_per_table;
  const int row_base = t << 4;

  const float* __restrict__ X = table ? x_dst : x_src;
  const float* __restrict__ w = weight + table * HIDDEN;

  const int m  = lane & 15;          // row within tile (A) / column N (B)
  const int kp = (lane >> 4) << 1;   // K sub-offset: 0 or 2

  int row = row_base + m;
  if (row >= n_nodes) row = n_nodes - 1;   // clamp: keep EXEC all-ones for WMMA
  const float* __restrict__ rp = X + (size_t)row * HIDDEN + kp;

  const float bmask = (m == 0) ? 1.0f : 0.0f;  // weight lives in column N==0

  v8f acc = {0.f, 0.f, 0.f, 0.f, 0.f, 0.f, 0.f, 0.f};
#pragma unroll
  for (int k0 = 0; k0 < HIDDEN; k0 += 4) {
    v2f a = *(const v2f*)(rp + k0);          // A: K = k0+kp, k0+kp+1
    v2f b;
    b.x = bmask * w[k0 + kp];                // B: K = k0+kp,  N = m
    b.y = bmask * w[k0 + kp + 1];            // B: K = k0+kp+1
    acc = __builtin_amdgcn_wmma_f32_16x16x4_f32(
        /*neg_a=*/false, a, /*neg_b=*/false, b,
        /*c_mod=*/(short)0, acc, /*reuse_a=*/false, /*reuse_b=*/false);
  }

  // D layout: VGPR r -> (M=r, N=lane) for lanes 0-15, (M=r+8, N=lane-16) for 16-31.
  // Column N==0 lives in lane 0 (rows 0..7) and lane 16 (rows 8..15).
  if (m == 0) {
    const int moff = (lane >> 4) << 3;  // 0 or 8
    float* __restrict__ s = s_out + (size_t)table * n_nodes + row_base + moff;
#pragma unroll
    for (int r = 0; r < 8; ++r) {
      if (row_base + moff + r < n_nodes) s[r] = acc[r];
    }
  }
}

// Pass 2: pure gather-add over L2-resident 400KB score tables.
// 4 edges/thread with 128-bit index loads and a 128-bit store.
__global__ __launch_bounds__(256) void edge_decode(
    const int* __restrict__ idx,        // [2 * n_edges] (int32)
    const float* __restrict__ s,        // [2 * n_nodes] precomputed scores
    const float* __restrict__ bias,     // [1]
    float* __restrict__ out,            // [n_edges]
    int n_edges, int n_nodes)
{
  const float b = bias[0];
  const float* __restrict__ s_src = s;
  const float* __restrict__ s_dst = s + n_nodes;

  const int e4 = (blockIdx.x * blockDim.x + threadIdx.x) << 2;
  if (e4 + 3 < n_edges) {
    int4 i0 = *(const int4*)(idx + e4);
    int4 i1 = *(const int4*)(idx + n_edges + e4);
    float4 r;
    r.x = s_src[i0.x] + s_dst[i1.x] + b;
    r.y = s_src[i0.y] + s_dst[i1.y] + b;
    r.z = s_src[i0.z] + s_dst[i1.z] + b;
    r.w = s_src[i0.w] + s_dst[i1.w] + b;
    *(float4*)(out + e4) = r;
  } else {
    for (int e = e4; e < n_edges; ++e)
      out[e] = s_src[idx[e]] + s_dst[idx[n_edges + e]] + b;
  }
}

extern "C" void kernel_launch(void* const* d_in, const int* in_sizes, int n_in,
                              void* d_out, int out_size, void* d_ws, size_t ws_size,
                              hipStream_t stream)
{
  const float* x_src = (const float*)d_in[0];   // [n_nodes, 128]
  const float* x_dst = (const float*)d_in[1];   // [n_nodes, 128]
  const int*   idx   = (const int*)  d_in[2];   // [2, n_edges]
  const float* wgt   = (const float*)d_in[3];   // [1, 256]
  const float* bias  = (const float*)d_in[4];   // [1]
  float* out = (float*)d_out;
  float* s   = (float*)d_ws;                    // 2*n_nodes floats (800 KB)

  const int n_nodes = in_sizes[0] / HIDDEN;
  const int n_edges = in_sizes[2] / 2;

  // Pass 1: WMMA GEMV precompute. 8 waves (tiles) per 256-thread block.
  const int tiles_per_table = (n_nodes + 15) / 16;
  const int total_tiles = 2 * tiles_per_table;
  const int pre_blocks  = (total_tiles + 7) / 8;
  gemv_precompute<<<pre_blocks, 256, 0, stream>>>(x_src, x_dst, wgt, s,
                                                  n_nodes, tiles_per_table);

  // Pass 2: edge gather-add (stream-ordered after pass 1).
  const int threads   = (n_edges + 3) / 4;
  const int ed_blocks = (threads + 255) / 256;
  edge_decode<<<ed_blocks, 256, 0, stream>>>(idx, s, bias, out,
                                             n_edges, n_nodes);
}